// SAGEConvModel_35278861369955
// MI455X (gfx1250) — compile-verified
//
#include <hip/hip_runtime.h>

// GraphSAGE 3-layer forward for MI455X (gfx1250, wave32).
// Aggregation is L2-resident atomic scatter; GEMMs use V_WMMA_F32_16X16X4_F32
// with K-pair-swizzled weights in LDS so each B fragment is one ds_load_b64.

typedef __attribute__((ext_vector_type(2))) float v2f;
typedef __attribute__((ext_vector_type(8))) float v8f;

#define D_H 128

// ---- degree ---------------------------------------------------------------
__global__ void k_deg(const int* __restrict__ dst, float* __restrict__ cnt, int nE) {
  int i = blockIdx.x * blockDim.x + threadIdx.x;
  if (i < nE) atomicAdd(&cnt[dst[i]], 1.0f);
}

__global__ void k_invdeg(const float* __restrict__ cnt, float* __restrict__ inv, int n) {
  int i = blockIdx.x * blockDim.x + threadIdx.x;
  if (i < n) inv[i] = 1.0f / fmaxf(cnt[i], 1.0f);
}

// ---- h0 = x[:, 4:10] ------------------------------------------------------
__global__ void k_slice(const float* __restrict__ x, float* __restrict__ h0, int n) {
  int i = blockIdx.x * blockDim.x + threadIdx.x;
  if (i < n * 6) {
    int node = i / 6, f = i - node * 6;
    h0[i] = x[node * 10 + 4 + f];
  }
}

// ---- edge-parallel aggregation, 6-wide ------------------------------------
__global__ void k_agg6(const int* __restrict__ src, const int* __restrict__ dst,
                       const float* __restrict__ h, float* __restrict__ sums, int nE) {
  int e = blockIdx.x * blockDim.x + threadIdx.x;
  if (e >= nE) return;
  int s = src[e], d = dst[e];
#pragma unroll
  for (int f = 0; f < 6; ++f) atomicAdd(&sums[d * 6 + f], h[s * 6 + f]);
}

// ---- edge-parallel aggregation, 128-wide (float4 per thread) --------------
__global__ void k_agg128(const int* __restrict__ src, const int* __restrict__ dst,
                         const float* __restrict__ h, float* __restrict__ sums, int nE) {
  long long i = (long long)blockIdx.x * blockDim.x + threadIdx.x;
  int e = (int)(i >> 5);
  if (e >= nE) return;
  int g = (int)(i & 31);
  int s = src[e], d = dst[e];
  float4 v = *(const float4*)(h + (size_t)s * D_H + g * 4);
  float* p = sums + (size_t)d * D_H + g * 4;
  atomicAdd(p + 0, v.x);
  atomicAdd(p + 1, v.y);
  atomicAdd(p + 2, v.z);
  atomicAdd(p + 3, v.w);
}

// ---- layer0: K=6, N=128; plain VALU (too small for WMMA) ------------------
__global__ void k_layer0(const float* __restrict__ sum0, const float* __restrict__ h0,
                         const float* __restrict__ inv, const float* __restrict__ Wl,
                         const float* __restrict__ Wr, const float* __restrict__ b,
                         float* __restrict__ out, int n) {
  long long i = (long long)blockIdx.x * blockDim.x + threadIdx.x;
  if (i >= (long long)n * 128) return;
  int node = (int)(i >> 7), o = (int)(i & 127);
  float s = inv[node];
  float acc = b[o];
#pragma unroll
  for (int f = 0; f < 6; ++f)
    acc += sum0[node * 6 + f] * s * Wl[f * 128 + o] + h0[node * 6 + f] * Wr[f * 128 + o];
  out[i] = fmaxf(acc, 0.0f);
}

// ---- layers 1/2: out = relu(mean@Wl + h@Wr + b), K=128, N = NT*16 ---------
// One wave owns a 16-row M tile and all NT column tiles.
// A fragment (16x4 f32): lane<16 -> row=lane&15, K={k,k+1}; lane>=16 -> K={k+2,k+3}.
// B fragment (4x16 f32): lane<16 -> K rows {k,k+1}; lanes>=16 -> {k+2,k+3}; N=lane&15.
// W is staged in LDS as K-pairs: lds2[(k4*N + n)*2 + kk] = (W[4k4+2kk][n], W[4k4+2kk+1][n])
// so each lane's B fragment is a single aligned ds_load_b64 (conflict-free).
template <int NT>
__global__ __launch_bounds__(128) void k_gemm_wmma(
    const float* __restrict__ sums, const float* __restrict__ h,
    const float* __restrict__ inv, const float* __restrict__ Wl,
    const float* __restrict__ Wr, const float* __restrict__ bias,
    float* __restrict__ out, int n, int n_tiles) {
  constexpr int N = NT * 16;
  constexpr int K = 128;
  __shared__ __align__(16) v2f lds2[(K / 2) * N];  // <= 64 KB; Wl then Wr (two passes)

  const int tid = threadIdx.x;
  const int wave = tid >> 5, lane = tid & 31;
  const int tile = blockIdx.x * 4 + wave;          // wave-uniform
  const bool valid = tile < n_tiles;               // wave-uniform -> EXEC all-1s at WMMA
  const int col = lane & 15;
  const int kk = lane >> 4;                        // half-wave K sub-group
  const int koff = kk * 2;
  int row = tile * 16 + col;
  const int rowc = (row < n) ? row : (n - 1);      // clamp A reads (pad rows)
  float s = 0.0f;
  if (valid) s = inv[rowc];

  v8f acc[NT] = {};
  const float* Wsrc[2] = {Wl, Wr};
  const float* Asrc[2] = {sums, h};

  for (int pass = 0; pass < 2; ++pass) {
    __syncthreads();
    {
      const float* Wp = Wsrc[pass];
      for (int j = tid; j < (K / 2) * N; j += 128) {
        int k2 = j / N;             // K-pair index 0..63
        int nn = j - k2 * N;        // column
        v2f pr;
        pr.x = Wp[(2 * k2 + 0) * N + nn];
        pr.y = Wp[(2 * k2 + 1) * N + nn];
        lds2[((k2 >> 1) * N + nn) * 2 + (k2 & 1)] = pr;
      }
    }
    __syncthreads();
    if (valid) {
      const float* A = Asrc[pass] + (size_t)rowc * K + koff;
#pragma unroll 4
      for (int k = 0; k < K; k += 4) {
        v2f a = *(const v2f*)(A + k);
        if (pass == 0) { a.x *= s; a.y *= s; }   // fold mean = sums * invdeg
        const v2f* bbase = lds2 + ((k >> 2) * N + col) * 2 + kk;
#pragma unroll
        for (int t = 0; t < NT; ++t) {
          v2f b = bbase[t * 32];                 // ds_load_b64
          acc[t] = __builtin_amdgcn_wmma_f32_16x16x4_f32(
              false, a, false, b, (short)0, acc[t], false, false);
        }
      }
    }
  }

  if (valid) {
    const int mbase = kk * 8;  // C/D layout: lanes 16-31 hold M=8..15
#pragma unroll
    for (int t = 0; t < NT; ++t) {
#pragma unroll
      for (int r = 0; r < 8; ++r) {
        int gr = tile * 16 + mbase + r;
        if (gr < n) {
          float v = acc[t][r] + bias[t * 16 + col];
          out[(size_t)gr * N + t * 16 + col] = fmaxf(v, 0.0f);
        }
      }
    }
  }
}

// ---------------------------------------------------------------------------
extern "C" void kernel_launch(void* const* d_in, const int* in_sizes, int n_in,
                              void* d_out, int out_size, void* d_ws, size_t ws_size,
                              hipStream_t stream) {
  const float* x   = (const float*)d_in[0];
  const int*   ei  = (const int*)d_in[1];
  const float* Wl0 = (const float*)d_in[2];
  const float* bl0 = (const float*)d_in[3];
  const float* Wr0 = (const float*)d_in[4];
  const float* Wl1 = (const float*)d_in[5];
  const float* bl1 = (const float*)d_in[6];
  const float* Wr1 = (const float*)d_in[7];
  const float* Wl2 = (const float*)d_in[8];
  const float* bl2 = (const float*)d_in[9];
  const float* Wr2 = (const float*)d_in[10];

  const int n  = in_sizes[0] / 10;   // 100000
  const int nE = in_sizes[1] / 2;    // 1600000
  const int* src  = ei;
  const int* dstA = ei + nE;

  // scratch carve-out (256B aligned slices)
  char* ws = (char*)d_ws;
  size_t off = 0;
  auto carve = [&](size_t bytes) -> float* {
    float* p = (float*)(ws + off);
    off += (bytes + 255) & ~(size_t)255;
    return p;
  };
  float* cnt  = carve((size_t)n * 4);
  float* inv  = carve((size_t)n * 4);
  float* h0   = carve((size_t)n * 6 * 4);
  float* sum0 = carve((size_t)n * 6 * 4);
  float* h1   = carve((size_t)n * D_H * 4);
  float* sums = carve((size_t)n * D_H * 4);
  float* h2   = carve((size_t)n * D_H * 4);
  (void)ws_size;

  const int TPB = 256;
  const int n_tiles = (n + 15) / 16;
  const int gemm_blocks = (n_tiles + 3) / 4;
  const int agg128_blocks = (int)(((long long)nE * 32 + TPB - 1) / TPB);

  // degrees (shared by all layers)
  hipMemsetAsync(cnt, 0, (size_t)n * 4, stream);
  k_deg<<<(nE + TPB - 1) / TPB, TPB, 0, stream>>>(dstA, cnt, nE);
  k_invdeg<<<(n + TPB - 1) / TPB, TPB, 0, stream>>>(cnt, inv, n);

  // layer 0
  k_slice<<<(n * 6 + TPB - 1) / TPB, TPB, 0, stream>>>(x, h0, n);
  hipMemsetAsync(sum0, 0, (size_t)n * 6 * 4, stream);
  k_agg6<<<(nE + TPB - 1) / TPB, TPB, 0, stream>>>(src, dstA, h0, sum0, nE);
  k_layer0<<<(int)(((long long)n * 128 + TPB - 1) / TPB), TPB, 0, stream>>>(
      sum0, h0, inv, Wl0, Wr0, bl0, h1, n);

  // layer 1 (WMMA, N=128)
  hipMemsetAsync(sums, 0, (size_t)n * D_H * 4, stream);
  k_agg128<<<agg128_blocks, TPB, 0, stream>>>(src, dstA, h1, sums, nE);
  k_gemm_wmma<8><<<gemm_blocks, 128, 0, stream>>>(sums, h1, inv, Wl1, Wr1, bl1, h2, n, n_tiles);

  // layer 2 (WMMA, N=64) -> d_out
  hipMemsetAsync(sums, 0, (size_t)n * D_H * 4, stream);
  k_agg128<<<agg128_blocks, TPB, 0, stream>>>(src, dstA, h2, sums, nE);
  k_gemm_wmma<4><<<gemm_blocks, 128, 0, stream>>>(sums, h2, inv, Wl2, Wr2, bl2,
                                                  (float*)d_out, n, n_tiles);
}